// GaussianVoxelizer_78426102825218
// MI455X (gfx1250) — compile-verified
//
#include <hip/hip_runtime.h>
#include <hip/hip_bf16.h>

// ---------------------------------------------------------------------------
// Gaussian voxelizer, gather formulation, WMMA f32 16x16x4 feature accumulate.
// Grid: 200 x 200 x 16, voxel 0.4, footprint +/-2 voxels, D=16 features.
// ---------------------------------------------------------------------------

#define N_GAUSS   60000
#define GX        200
#define GY        200
#define GZ        16
#define NVOX      (GX * GY * GZ)      // 640000
#define DFEAT     16
#define BINS_X    201                  // center voxel x can be 0..200
#define BINS_Y    201
#define NBINS     (BINS_X * BINS_Y)
#define BIN_CAP   32
#define VMINX     (-40.0f)
#define VMINY     (-40.0f)
#define VMINZ     (-1.0f)
#define VMAXX     (40.0f)
#define VMAXY     (40.0f)
#define VMAXZ     (5.4f)
#define VOXSZ     (0.4f)
#define EPSV      (1e-6f)

// per-wave LDS candidate staging: 25 bins * 32 cap + 3 sentinels, rounded up
#define CAND_MAX  804
#define WAVES_PER_BLOCK 8

typedef __attribute__((ext_vector_type(2))) float v2f;
typedef __attribute__((ext_vector_type(8))) float v8f;

// ---------------------------------------------------------------------------
// Kernel 1: zero bin counters
// ---------------------------------------------------------------------------
__global__ void vox_zero_bins(int* __restrict__ binCount) {
    int i = blockIdx.x * blockDim.x + threadIdx.x;
    if (i < NBINS) binCount[i] = 0;
}

// ---------------------------------------------------------------------------
// Kernel 2: per-gaussian preprocess + binning
// Packed layout (4 float4 per gaussian):
//   p0 = (mx, my, mz, opacity_eff)           opacity_eff = 0 if invalid
//   p1 = (i00, i01s, i02s, i11)              symmetrized inverse covariance
//   p2 = (i12s, i22, r0, r1)                 radii = 3*sqrt(diag(cov))
//   p3 = (r2, czf, 0, 0)                     czf = floor((mz-vminz)/vox)
// ---------------------------------------------------------------------------
__global__ void vox_prep(const float* __restrict__ means,
                         const float* __restrict__ cov,
                         const float* __restrict__ opac,
                         float4* __restrict__ P,
                         int* __restrict__ binCount,
                         int* __restrict__ binIdx) {
    int g = blockIdx.x * blockDim.x + threadIdx.x;
    if (g >= N_GAUSS) return;

    float mx = means[g * 3 + 0];
    float my = means[g * 3 + 1];
    float mz = means[g * 3 + 2];
    const float* c = cov + g * 9;
    float a00 = c[0], a01 = c[1], a02 = c[2];
    float a10 = c[3], a11 = c[4], a12 = c[5];
    float a20 = c[6], a21 = c[7], a22 = c[8];

    float det = a00 * (a11 * a22 - a12 * a21)
              - a01 * (a10 * a22 - a12 * a20)
              + a02 * (a10 * a21 - a11 * a20);
    float rd = 1.0f / det;
    float i00 = (a11 * a22 - a12 * a21) * rd;
    float i01 = (a02 * a21 - a01 * a22) * rd;
    float i02 = (a01 * a12 - a02 * a11) * rd;
    float i10 = (a12 * a20 - a10 * a22) * rd;
    float i11 = (a00 * a22 - a02 * a20) * rd;
    float i12 = (a02 * a10 - a00 * a12) * rd;
    float i20 = (a10 * a21 - a11 * a20) * rd;
    float i21 = (a01 * a20 - a00 * a21) * rd;
    float i22 = (a00 * a11 - a01 * a10) * rd;
    // d^T M d only sees the symmetric part of M
    float i01s = 0.5f * (i01 + i10);
    float i02s = 0.5f * (i02 + i20);
    float i12s = 0.5f * (i12 + i21);

    float op = opac[g];
    bool valid = (op > EPSV) &&
                 (mx >= VMINX) && (mx <= VMAXX) &&
                 (my >= VMINY) && (my <= VMAXY) &&
                 (mz >= VMINZ) && (mz <= VMAXZ);

    float cxf = floorf((mx - VMINX) / VOXSZ);
    float cyf = floorf((my - VMINY) / VOXSZ);
    float czf = floorf((mz - VMINZ) / VOXSZ);

    float r0 = 3.0f * sqrtf(a00);
    float r1 = 3.0f * sqrtf(a11);
    float r2 = 3.0f * sqrtf(a22);

    P[g * 4 + 0] = make_float4(mx, my, mz, valid ? op : 0.0f);
    P[g * 4 + 1] = make_float4(i00, i01s, i02s, i11);
    P[g * 4 + 2] = make_float4(i12s, i22, r0, r1);
    P[g * 4 + 3] = make_float4(r2, czf, 0.0f, 0.0f);

    if (valid) {
        int cx = (int)cxf;
        int cy = (int)cyf;
        if (cx >= 0 && cx < BINS_X && cy >= 0 && cy < BINS_Y) {
            int bin = cx * BINS_Y + cy;
            int slot = atomicAdd(&binCount[bin], 1);
            if (slot < BIN_CAP) binIdx[bin * BIN_CAP + slot] = g;
        }
    }
}

// ---------------------------------------------------------------------------
// Kernel 3: sort each bin by gaussian index (determinism: atomic fill order
// must not influence FP accumulation order). Bins hold ~1.5 entries on avg.
// ---------------------------------------------------------------------------
__global__ void vox_sort_bins(int* __restrict__ binCount, int* __restrict__ binIdx) {
    int b = blockIdx.x * blockDim.x + threadIdx.x;
    if (b >= NBINS) return;
    int n = binCount[b];
    if (n > BIN_CAP) { n = BIN_CAP; binCount[b] = BIN_CAP; }
    int* p = binIdx + b * BIN_CAP;
    for (int i = 1; i < n; ++i) {
        int v = p[i];
        int j = i - 1;
        while (j >= 0 && p[j] > v) { p[j + 1] = p[j]; --j; }
        p[j + 1] = v;
    }
}

// ---------------------------------------------------------------------------
// Main kernel helpers
// ---------------------------------------------------------------------------
__device__ __forceinline__ float vox_eval_w(const float4* __restrict__ P, int g,
                                            float vx, float vy, float vz, float zf) {
    int gg = g < 0 ? 0 : g;
    float4 p0 = P[gg * 4 + 0];
    float4 p1 = P[gg * 4 + 1];
    float4 p2 = P[gg * 4 + 2];
    float4 p3 = P[gg * 4 + 3];
    float dx = vx - p0.x;
    float dy = vy - p0.y;
    float dz = vz - p0.z;
    bool ok = (g >= 0)
           && (fabsf(zf - p3.y) <= 2.0f)      // z footprint window
           && (fabsf(dx) <= p2.z)             // ellipsoid bbox test
           && (fabsf(dy) <= p2.w)
           && (fabsf(dz) <= p3.x);
    float q = p1.x * dx * dx + p1.w * dy * dy + p2.y * dz * dz
            + 2.0f * (p1.y * dx * dy + p1.z * dx * dz + p2.x * dy * dz);
    return ok ? p0.w * __expf(-0.5f * q) : 0.0f;
}

// One K=4 WMMA chunk: A = w[16 voxels x 4 gaussians], B = F[4 gaussians x 16].
// A layout (f32 16x4): lanes 0-15 -> K=0,1 in the two A regs; lanes 16-31 -> K=2,3.
// B layout (f32 4x16): same K pattern, N striped across lanes within each half.
__device__ __forceinline__ void vox_chunk(v8f& c, float& dens,
                                          const float4* __restrict__ P,
                                          const float* __restrict__ feats,
                                          int g0, int g1, int g2, int g3,
                                          int h, int f,
                                          float vx, float vy, float vz, float zf) {
    int ga = h ? g2 : g0;          // this half's K-slots: 2h, 2h+1
    int gb = h ? g3 : g1;
    float wa = vox_eval_w(P, ga, vx, vy, vz, zf);
    float wb = vox_eval_w(P, gb, vx, vy, vz, zf);
    v2f A; A.x = wa; A.y = wb;
    int fa = ga < 0 ? 0 : ga;
    int fb = gb < 0 ? 0 : gb;
    v2f B;
    B.x = feats[fa * DFEAT + f];
    B.y = feats[fb * DFEAT + f];
    // D = A * B + C   (emits v_wmma_f32_16x16x4_f32)
    c = __builtin_amdgcn_wmma_f32_16x16x4_f32(false, A, false, B, (short)0, c,
                                              false, false);
    dens += wa + wb;
}

// ---------------------------------------------------------------------------
// Kernel 4: one wave32 per z-column (x,y).
// Phase 1: parallel gather of candidate gaussian indices (one neighbor bin
//          per lane, wave prefix-sum for compaction) into per-wave LDS list.
// Phase 2: scalar-uniform chunk loop: 4 candidates per V_WMMA_F32_16X16X4_F32,
//          accumulating the [16 voxel x 16 feature] tile in the C operand.
// Density via per-lane sums + cross-half shuffle. Each voxel written once.
// ---------------------------------------------------------------------------
__global__ __launch_bounds__(256) void vox_main(const float4* __restrict__ P,
                                                const int* __restrict__ binCount,
                                                const int* __restrict__ binIdx,
                                                const float* __restrict__ feats,
                                                float* __restrict__ outDensity,
                                                float* __restrict__ outFeats) {
    __shared__ int s_cand[WAVES_PER_BLOCK][CAND_MAX];

    int tid  = blockIdx.x * blockDim.x + threadIdx.x;
    int wave = tid >> 5;                        // global column id (exact grid)
    int wib  = (threadIdx.x >> 5);              // wave in block
    int lane = threadIdx.x & 31;
    int x = wave / GY;
    int y = wave % GY;
    int h = lane >> 4;                 // half-wave id
    int f = lane & 15;                 // feature column (also voxel z for W eval)
    float zf = (float)f;
    float vx = VMINX + ((float)x + 0.5f) * VOXSZ;
    float vy = VMINY + ((float)y + 0.5f) * VOXSZ;
    float vz = VMINZ + (zf + 0.5f) * VOXSZ;

    // ---------------- Phase 1: gather candidates into LDS ----------------
    int bx0 = max(x - 2, 0), bx1 = min(x + 2, BINS_X - 1);
    int by0 = max(y - 2, 0), by1 = min(y + 2, BINS_Y - 1);
    int nbx = bx1 - bx0 + 1;
    int nby = by1 - by0 + 1;
    int nbinsw = nbx * nby;                     // <= 25

    int myBin = 0, myCnt = 0;
    if (lane < nbinsw) {
        int bx = bx0 + lane / nby;
        int by = by0 + lane % nby;
        myBin = bx * BINS_Y + by;
        int c = binCount[myBin];
        myCnt = c > BIN_CAP ? BIN_CAP : c;
    }
    // inclusive wave scan of myCnt
    int incl = myCnt;
#pragma unroll
    for (int d = 1; d < 32; d <<= 1) {
        int t = __shfl_up(incl, d, 32);
        if (lane >= d) incl += t;
    }
    int excl  = incl - myCnt;
    int total = __shfl(incl, 31, 32);

    for (int i = 0; i < myCnt; ++i) {
        s_cand[wib][excl + i] = binIdx[myBin * BIN_CAP + i];
    }
    if (lane < 3) s_cand[wib][total + lane] = -1;   // sentinels for tail chunk

    __syncthreads();   // uniform: every thread reaches this exactly once

    // ---------------- Phase 2: uniform WMMA chunk loop ----------------
    v8f c = {};
    float dens = 0.0f;

    int total_s = __builtin_amdgcn_readfirstlane(total);
    for (int i = 0; i < total_s; i += 4) {
        int g0 = s_cand[wib][i + 0];            // LDS broadcast reads
        int g1 = s_cand[wib][i + 1];
        int g2 = s_cand[wib][i + 2];
        int g3 = s_cand[wib][i + 3];
        vox_chunk(c, dens, P, feats, g0, g1, g2, g3, h, f, vx, vy, vz, zf);
    }

    // density[z = lane&15] = this-half partial + other-half partial
    float dtot = dens + __shfl_xor(dens, 16, 32);

    int colBase = x * (GY * GZ) + y * GZ;      // flat voxel base of this column
    if (lane < 16) {
        outDensity[colBase + f] = dtot;
    }

    // C element (row m, col lane) = feats tile at voxel z = m + 8*h, feature f.
    // Density for that z lives on lane (m + 8*h); fetch via lane shuffle.
#pragma unroll
    for (int m = 0; m < 8; ++m) {
        int z = m + 8 * h;
        float dsel = __shfl(dtot, m + 8 * h, 32);
        outFeats[(size_t)(colBase + z) * DFEAT + f] = c[m] / fmaxf(dsel, EPSV);
    }
}

// ---------------------------------------------------------------------------
// Host launcher
// ---------------------------------------------------------------------------
extern "C" void kernel_launch(void* const* d_in, const int* in_sizes, int n_in,
                              void* d_out, int out_size, void* d_ws, size_t ws_size,
                              hipStream_t stream) {
    const float* means = (const float*)d_in[0];   // [N,3]
    const float* cov   = (const float*)d_in[1];   // [N,3,3]
    const float* opac  = (const float*)d_in[2];   // [N]
    const float* feats = (const float*)d_in[3];   // [N,16]

    float* outDensity = (float*)d_out;            // [640000]
    float* outFeats   = outDensity + NVOX;        // [640000*16]

    // Workspace layout (256B-aligned offsets):
    //   P         : N_GAUSS * 4 * float4 = 3,840,000 B
    //   binCount  : NBINS ints           =   161,604 B
    //   binIdx    : NBINS * BIN_CAP ints = 5,171,328 B
    char* ws = (char*)d_ws;
    float4* P     = (float4*)(ws);
    int* binCount = (int*)(ws + 3840000);
    int* binIdx   = (int*)(ws + 4001792);

    vox_zero_bins<<<(NBINS + 255) / 256, 256, 0, stream>>>(binCount);
    vox_prep<<<(N_GAUSS + 255) / 256, 256, 0, stream>>>(means, cov, opac,
                                                        P, binCount, binIdx);
    vox_sort_bins<<<(NBINS + 255) / 256, 256, 0, stream>>>(binCount, binIdx);

    // 40000 columns, 8 waves (256 threads) per block -> exactly 5000 blocks
    vox_main<<<(GX * GY) / 8, 256, 0, stream>>>(P, binCount, binIdx, feats,
                                                outDensity, outFeats);
}